// SphericalBasisLayer_30408368456387
// MI455X (gfx1250) — compile-verified
//
#include <hip/hip_runtime.h>

#define NUM_SPH 7
#define NUM_RAD 6
#define NUM_C   42        // 7*6

typedef __attribute__((ext_vector_type(2))) float v2f;
typedef __attribute__((ext_vector_type(8))) float v8f;

// pref[l] = sqrt((2l+1)/(4*pi))
#define PREF0 0.28209479177387814f
#define PREF1 0.48860251190291992f
#define PREF2 0.63078313050504009f
#define PREF3 0.74635266518023113f
#define PREF4 0.84628437532163443f
#define PREF5 0.93560257962738880f
#define PREF6 1.01710723628205458f

// coeff[k][l] = monomial coefficient of ct^k in pref[l]*P_l(ct); row 7 = zero pad.
__constant__ float c_coeff[8][8] = {
  { PREF0, 0.f,   -0.5f*PREF2, 0.f,         0.375f*PREF4, 0.f,          -0.3125f*PREF6,  0.f },
  { 0.f,   PREF1, 0.f,         -1.5f*PREF3, 0.f,          1.875f*PREF5,  0.f,            0.f },
  { 0.f,   0.f,   1.5f*PREF2,  0.f,         -3.75f*PREF4, 0.f,           6.5625f*PREF6,  0.f },
  { 0.f,   0.f,   0.f,         2.5f*PREF3,  0.f,          -8.75f*PREF5,  0.f,            0.f },
  { 0.f,   0.f,   0.f,         0.f,         4.375f*PREF4, 0.f,           -19.6875f*PREF6,0.f },
  { 0.f,   0.f,   0.f,         0.f,         0.f,          7.875f*PREF5,  0.f,            0.f },
  { 0.f,   0.f,   0.f,         0.f,         0.f,          0.f,           14.4375f*PREF6, 0.f },
  { 0.f,   0.f,   0.f,         0.f,         0.f,          0.f,           0.f,            0.f }
};

// ---------------- Kernel A: rbf table, one thread per (edge, l*6+k) ----------------
__global__ void rbf_table_kernel(const float* __restrict__ dist,
                                 const float* __restrict__ zeros,
                                 const float* __restrict__ norm,
                                 float* __restrict__ rbf, int total) {
  int g = blockIdx.x * blockDim.x + threadIdx.x;
  if (g >= total) return;
  int e = g / NUM_C;
  int c = g - e * NUM_C;
  int l = c / NUM_RAD;

  float x = dist[e] * 0.2f;                 // dist / CUTOFF
  // envelope, p = 6:  1/x - 28 x^5 + 48 x^6 - 21 x^7
  float x2 = x * x;
  float x5 = x2 * x2 * x;
  float env = 1.0f / x + x5 * (-28.0f + x * (48.0f - 21.0f * x));

  float arg = zeros[c] * x;
  float s, co;
  sincosf(arg, &s, &co);
  float inv = 1.0f / arg;
  float j = s * inv;                        // j0
  if (l > 0) {
    float jm = j;
    float jc = (s * inv - co) * inv;        // j1
    #pragma unroll 1
    for (int ll = 2; ll <= l; ++ll) {
      float t = (2.0f * (float)ll - 1.0f) * inv * jc - jm;
      jm = jc; jc = t;
    }
    j = jc;
  }
  rbf[g] = env * norm[c] * j;
}

// ---------------- Kernel B: gather * cbf, 16 triplets per wave ----------------
// cbf for 16 triplets computed with two V_WMMA_F32_16X16X4_F32 ops:
//   D(16x16) = [ct^k] (16x4) x coeff (4x16), accumulated over K-tiles {0..3},{4..7}.
__global__ void __launch_bounds__(256)
triplet_kernel(const float* __restrict__ angle,
               const int*   __restrict__ idx_kj,
               const float* __restrict__ rbf,
               float*       __restrict__ out, int T_) {
  const int lane = threadIdx.x & 31;
  const int wave = threadIdx.x >> 5;

  __shared__ float s_cbf[8][16][8];   // [wave][triplet m][l], padded to 8
  __shared__ int   s_idx[8][16];

  const int t0 = (blockIdx.x * (blockDim.x >> 5) + wave) * 16;
  if (t0 >= T_) return;               // wave-uniform exit; EXEC stays full below
  const int jmax = min(16, T_ - t0);  // wave-uniform trip count for the store loop

  // lanes 0..15: per-triplet scalar work (cos + idx load + row prefetch)
  float ct = 0.0f;
  if (lane < 16) {
    int t = t0 + lane;
    int idx = 0;
    if (t < T_) {
      ct  = cosf(angle[t]);
      idx = idx_kj[t];
      const float* row = rbf + idx * NUM_C;
      __builtin_prefetch(row, 0, 1);       // global_prefetch_b8 (row spans 2-3 lines)
      __builtin_prefetch(row + 32, 0, 1);
    }
    s_idx[wave][lane] = idx;
  }

  // Broadcast ct of triplet m = lane%16 to both half-waves, build monomials.
  float ctm = __shfl(ct, lane & 15);
  float c1 = ctm, c2 = ctm * ctm;
  float c3 = c2 * ctm, c4 = c2 * c2, c5 = c4 * ctm, c6 = c3 * c3;

  const int half = lane >> 4;         // K group: lanes 0-15 -> even K base, 16-31 -> +2
  const int n    = lane & 15;         // B/D column

  // A tile layout (f32 16x4): v0 = K=2*half, v1 = K=2*half+1
  v2f a0, a1, b0, b1;
  a0.x = half ? c2 : 1.0f;  a0.y = half ? c3 : c1;    // K tile 0..3
  a1.x = half ? c6 : c4;    a1.y = half ? 0.0f : c5;  // K tile 4..7 (row 7 coeff = 0)

  // B tile layout (f32 4x16): v0 = row K=2*half, v1 = row K=2*half+1, col n
  float b00 = 0.f, b01 = 0.f, b10 = 0.f, b11 = 0.f;
  if (n < NUM_SPH) {
    b00 = c_coeff[2 * half + 0][n];
    b01 = c_coeff[2 * half + 1][n];
    b10 = c_coeff[2 * half + 4][n];
    b11 = c_coeff[2 * half + 5][n];
  }
  b0.x = b00; b0.y = b01;
  b1.x = b10; b1.y = b11;

  v8f acc = {};
  acc = __builtin_amdgcn_wmma_f32_16x16x4_f32(false, a0, false, b0, (short)0, acc, false, false);
  acc = __builtin_amdgcn_wmma_f32_16x16x4_f32(false, a1, false, b1, (short)0, acc, false, false);

  // D layout: VGPR r, lane -> M = r + 8*half, N = n
  if (n < NUM_SPH) {
    #pragma unroll
    for (int r = 0; r < 8; ++r)
      s_cbf[wave][r + 8 * half][n] = acc[r];
  }

  // Store loop: lanes 0..20 each handle a v2f (two floats) of the 42-wide row.
  // All predicates hoisted: 'act' is loop-invariant, trip count is uniform.
  const int c = lane * 2;
  const int la = c / 6;
  const int lb = (c + 1) / 6;

  if (c < NUM_C) {
    const float* __restrict__ cb_row = &s_cbf[wave][0][0];
    float*       __restrict__ op     = out + (long long)t0 * NUM_C + c;
    #pragma unroll 1
    for (int jj = 0; jj < jmax; ++jj) {
      int base = s_idx[wave][jj] * NUM_C;
      v2f rv = *(const v2f*)(rbf + base + c);
      v2f o;
      o.x = rv.x * cb_row[jj * 8 + la];
      o.y = rv.y * cb_row[jj * 8 + lb];
      // streaming 336MB output: non-temporal so the L2 keeps the rbf table
      __builtin_nontemporal_store(o, (v2f*)op);
      op += NUM_C;
    }
  }
}

extern "C" void kernel_launch(void* const* d_in, const int* in_sizes, int n_in,
                              void* d_out, int out_size, void* d_ws, size_t ws_size,
                              hipStream_t stream) {
  const float* dist   = (const float*)d_in[0];
  const float* angle  = (const float*)d_in[1];
  const float* zeros  = (const float*)d_in[2];
  const float* norm   = (const float*)d_in[3];
  const int*   idx_kj = (const int*)d_in[4];
  const int E_ = in_sizes[0];
  const int T_ = in_sizes[1];

  float* rbf = (float*)d_ws;          // E * 42 floats (~67 MB, L2-resident)

  int totalA = E_ * NUM_C;
  rbf_table_kernel<<<(totalA + 255) / 256, 256, 0, stream>>>(dist, zeros, norm, rbf, totalA);

  // 256 threads = 8 waves = 128 triplets per block
  int blocks = (T_ + 127) / 128;
  triplet_kernel<<<blocks, 256, 0, stream>>>(angle, idx_kj, rbf, (float*)d_out, T_);
}